// Losses_4784593568314
// MI455X (gfx1250) — compile-verified
//
#include <hip/hip_runtime.h>
#include <hip/hip_bf16.h>

// ---------------------------------------------------------------------------
// Loss = ALPHA*SILog(output,depth) + BETA*(bins chamfer) [+ GAMMA*minmax]
// Chamfer pairwise distances computed on the CDNA5 matrix pipe:
//   (c - t)^2 = [c^2, -2c, 1, 0] . [1, t, t^2, 0]^T  ->  V_WMMA_F32_16X16X4_F32
// All min-reductions run in signed-int bit-space (exact for the nonnegative
// distances; order-independent atomics -> deterministic). Pixel chunk staged
// into LDS with GLOBAL_LOAD_ASYNC_TO_LDS_B32 (ASYNCcnt path).
// ---------------------------------------------------------------------------

typedef __attribute__((ext_vector_type(2))) float v2f;
typedef __attribute__((ext_vector_type(8))) float v8f;

#define EPSF      0.01f
#define LAMBF     0.85f
#define K_B       8
#define K_H       228
#define K_W       304
#define K_P       (K_H * K_W)      // 69312 pixels per batch
#define K_NB      256              // bins
#define K_CHUNK   2048             // pixels per workgroup
#define K_NPT     (K_CHUNK / 16)   // pixel tiles per chunk
#define K_CHUNKS  34               // 34*2048 = 69632 >= 69312
#define INF_BITS  0x7F800000       // +inf bit pattern (large positive int)

__device__ __forceinline__ int   imin2(int a, int b) { return a < b ? a : b; }
__device__ __forceinline__ int   imax2(int a, int b) { return a > b ? a : b; }

__device__ __forceinline__ float block_sum256(float v, float* red, int tid) {
    red[tid] = v;
    __syncthreads();
    #pragma unroll
    for (int s = 128; s > 0; s >>= 1) {
        if (tid < s) red[tid] += red[tid + s];
        __syncthreads();
    }
    float r = red[0];
    __syncthreads();
    return r;
}

__global__ __launch_bounds__(256) void loss_init(int* binmin, int* bmin, int* bmax) {
    int tid = threadIdx.x;
    for (int i = tid; i < K_B * K_NB; i += 256) binmin[i] = INF_BITS;
    if (tid < K_B) { bmin[tid] = INF_BITS; bmax[tid] = 0; }
}

__global__ __launch_bounds__(256) void loss_main(const float* __restrict__ outp,
                                                 const float* __restrict__ ctr,
                                                 const float* __restrict__ dep,
                                                 int* __restrict__ binmin,
                                                 int* __restrict__ bmin,
                                                 int* __restrict__ bmax,
                                                 float* __restrict__ parts) {
    __shared__ float s_ctr[K_NB];
    __shared__ float s_pix[K_CHUNK];
    __shared__ int   s_chamy[K_CHUNK];   // per-pixel min over all bins (bits)
    __shared__ int   s_binmin[K_NB];     // per-bin min over chunk pixels (bits)
    __shared__ float s_red[256];

    const int tid  = threadIdx.x;
    const int bid  = blockIdx.x;
    const int b    = bid / K_CHUNKS;
    const int ch   = bid % K_CHUNKS;
    const int base = ch * K_CHUNK;

    s_ctr[tid]    = ctr[b * K_NB + tid];
    s_binmin[tid] = INF_BITS;

    // ---- async stage of the pixel chunk: global -> LDS (ASYNCcnt path) ----
    for (int i = tid; i < K_CHUNK; i += 256) {
        const int idx = base + i;
        s_chamy[i] = INF_BITS;
        if (idx < K_P) {
            // LDS byte offset = low 32 bits of the flat shared address
            unsigned ldsoff = (unsigned)(size_t)&s_pix[i];
            const float* g  = dep + (size_t)b * K_P + idx;
            asm volatile("global_load_async_to_lds_b32 %0, %1, off"
                         :: "v"(ldsoff), "v"(g) : "memory");
        } else {
            s_pix[i] = -1.0f;   // padding -> invalid pixel
        }
    }
    asm volatile("s_wait_asynccnt 0x0" ::: "memory");
    __syncthreads();

    const int  lane = tid & 31;
    const int  wv   = tid >> 5;          // wave id 0..7
    const int  col  = lane & 15;         // matrix row/col owned by this lane
    const bool lo   = lane < 16;

    // Each wave owns 2 center-tiles (16 bins each) and sweeps all pixel tiles.
    #pragma unroll
    for (int cti = 0; cti < 2; ++cti) {
        const int ct = wv + cti * 8;           // center tile 0..15
        const float c = s_ctr[ct * 16 + col];
        v2f A;
        A.x = lo ? c * c      : 1.0f;          // K=0 / K=2 halves
        A.y = lo ? -2.0f * c  : 0.0f;          // K=1 / K=3 halves

        int accx[8];                            // running per-row (bin) min, bits
        #pragma unroll
        for (int j = 0; j < 8; ++j) accx[j] = INF_BITS;

        float t = s_pix[col];                   // software-pipelined load
        for (int pt = 0; pt < K_NPT; ++pt) {
            const float tnext = (pt + 1 < K_NPT) ? s_pix[(pt + 1) * 16 + col] : 0.0f;

            v2f Bv;
            Bv.x = lo ? 1.0f : t * t;
            Bv.y = lo ? t    : 0.0f;
            v8f Z = {};
            // D[m][n] = c_m^2 - 2 c_m t_n + t_n^2 = (c_m - t_n)^2
            v8f D = __builtin_amdgcn_wmma_f32_16x16x4_f32(
                        false, A, false, Bv, (short)0, Z, false, false);

            // column (pixel) validity: one select per tile
            const int vsel = (t >= EPSF) ? 0 : INF_BITS;
            int ymin = INF_BITS;
            #pragma unroll
            for (int j = 0; j < 8; ++j) {
                const int db = __float_as_int(D[j]);           // >= -eps -> signed
                ymin    = imin2(ymin, db);                     // min over rows
                accx[j] = imin2(accx[j], imax2(db, vsel));     // mask + min cols
            }
            // lanes L and L+16 (same pixel, complementary row halves) combine
            // in the LDS atomic unit -- no shuffle, no divergence.
            atomicMin(&s_chamy[pt * 16 + col], ymin);

            t = tnext;
        }

        // fold the 16 columns per row via same-address LDS atomic min
        #pragma unroll
        for (int j = 0; j < 8; ++j)
            atomicMin(&s_binmin[ct * 16 + j + ((lane >> 4) << 3)], accx[j]);
    }
    __syncthreads();

    // export per-bin minima (order-independent int atomic -> deterministic)
    atomicMin(&binmin[b * K_NB + tid], s_binmin[tid]);

    // pixel-wise pass: SILog partials, chamfer-y partials, depth min/max
    float sg = 0.0f, sg2 = 0.0f, ns = 0.0f, sumy = 0.0f, cnty = 0.0f;
    int dmn = INF_BITS, dmx = 0;
    for (int i = tid; i < K_CHUNK; i += 256) {
        const int idx = base + i;
        if (idx < K_P) {
            const float t  = s_pix[i];
            const int   tb = __float_as_int(t);     // depth >= 0
            dmn = imin2(dmn, tb);
            dmx = imax2(dmx, tb);
            const float o = outp[b * K_P + idx];
            if (o >= EPSF && t >= EPSF) {
                const float g = logf(o + EPSF) - logf(t + EPSF);
                sg += g; sg2 += g * g; ns += 1.0f;
            }
            if (t >= EPSF) {
                sumy += __int_as_float(s_chamy[i]);
                cnty += 1.0f;
            }
        }
    }
    atomicMin(&bmin[b], dmn);
    atomicMax(&bmax[b], dmx);

    const float r0 = block_sum256(sg,   s_red, tid);
    const float r1 = block_sum256(sg2,  s_red, tid);
    const float r2 = block_sum256(ns,   s_red, tid);
    const float r3 = block_sum256(sumy, s_red, tid);
    const float r4 = block_sum256(cnty, s_red, tid);
    if (tid == 0) {
        float* p = parts + bid * 5;
        p[0] = r0; p[1] = r1; p[2] = r2; p[3] = r3; p[4] = r4;
    }
}

__global__ __launch_bounds__(256) void loss_finalize(const int* __restrict__ epoch,
                                                     const float* __restrict__ ctr,
                                                     const int* __restrict__ binmin,
                                                     const int* __restrict__ bmin,
                                                     const int* __restrict__ bmax,
                                                     const float* __restrict__ parts,
                                                     float* __restrict__ outv) {
    __shared__ float s_red[256];
    __shared__ float s_chy[K_B];
    __shared__ float s_mm[K_B];
    const int tid = threadIdx.x;

    // cham_x: mean over all bins and batches = sum/(B*NB)
    float s = 0.0f;
    for (int i = tid; i < K_B * K_NB; i += 256)
        s += __int_as_float(binmin[i]);
    const float chamx = block_sum256(s, s_red, tid) / (float)(K_B * K_NB);

    if (tid < K_B) {
        float sy = 0.0f, cy = 0.0f;
        for (int c = 0; c < K_CHUNKS; ++c) {
            const float* p = parts + (tid * K_CHUNKS + c) * 5;
            sy += p[3]; cy += p[4];
        }
        s_chy[tid] = sy / cy;
        s_mm[tid]  = fabsf(ctr[tid * K_NB + (K_NB - 1)] - __int_as_float(bmax[tid]))
                   + fabsf(ctr[tid * K_NB]              - __int_as_float(bmin[tid]));
    }
    __syncthreads();

    if (tid == 0) {
        float sg = 0.0f, sg2 = 0.0f, n = 0.0f;
        for (int i = 0; i < K_B * K_CHUNKS; ++i) {
            const float* p = parts + i * 5;
            sg += p[0]; sg2 += p[1]; n += p[2];
        }
        const float mean = sg / n;
        const float var  = (sg2 - n * mean * mean) / (n - 1.0f);
        const float sil  = sqrtf(var + (1.0f - LAMBF) * mean * mean);

        float chy = 0.0f, mm = 0.0f;
        #pragma unroll
        for (int bb = 0; bb < K_B; ++bb) { chy += s_chy[bb]; mm += s_mm[bb]; }
        chy /= (float)K_B;

        float loss = 10.0f * sil + 0.1f * (chamx + chy);
        if (epoch[0] >= 10) loss += 0.1f * mm;
        outv[0] = loss;
    }
}

extern "C" void kernel_launch(void* const* d_in, const int* in_sizes, int n_in,
                              void* d_out, int out_size, void* d_ws, size_t ws_size,
                              hipStream_t stream) {
    const int*   epoch = (const int*)d_in[0];
    const float* outp  = (const float*)d_in[1];
    const float* ctr   = (const float*)d_in[2];
    const float* dep   = (const float*)d_in[3];
    // d_in[4] = lidar: dead code in the reference, unused.
    float* outv = (float*)d_out;

    int*   binmin = (int*)d_ws;
    int*   bmin   = binmin + K_B * K_NB;
    int*   bmax   = bmin + K_B;
    float* parts  = (float*)(bmax + K_B);

    loss_init<<<1, 256, 0, stream>>>(binmin, bmin, bmax);
    loss_main<<<K_B * K_CHUNKS, 256, 0, stream>>>(outp, ctr, dep,
                                                  binmin, bmin, bmax, parts);
    loss_finalize<<<1, 256, 0, stream>>>(epoch, ctr, binmin, bmin, bmax,
                                         parts, outv);
}